// TMMLayer_10728828305455
// MI455X (gfx1250) — compile-verified
//
#include <hip/hip_runtime.h>
#include <hip/hip_bf16.h>

// ---------------------------------------------------------------------------
// TMM (transfer-matrix method) forward pass for MI455X / gfx1250.
// One thread per (wavelength l, angle th) pair.
//
// Key transforms vs reference:
//  * cos(arcsin z) -> principal sqrt(1 - z^2); forward-angle fix = sign flip
//    of n*cos only (theta itself is never needed).
//  * Un-normalized interface matrices I~ = [[p+q, p-q],[p-q, p+q]]: scalar
//    prefactors cancel in r = A10/A00 and fold into t = 2^(NL+1)*prod(p_j)/A00.
//  * Only column 0 of the matrix product feeds the outputs, so the chain is
//    evaluated right-to-left as a 2-vector recurrence:
//        I~ * v = (u+w, u-w),  u = p*(v0+v1), w = q*(v0-v1)
//    i.e. 2 complex muls + 4 packed adds per interface (never forming I~),
//    plus 2 complex muls for the diag(e^{-i d}, e^{+i d}) propagation.
//    This halves the per-layer chain math vs the matrix-matrix form and
//    shrinks live state from a 2x2 complex matrix to a 2-vector.
//  * Complex mul = exactly 2 CDNA5 packed-FP32 VOP3P instructions via inline
//    asm (v_pk_mul_f32 + v_pk_fma_f32) with op_sel/op_sel_hi doing the
//    broadcast/swap and neg_lo the conjugate sign — zero shuffles.
//  * Per-l inputs indexed by uniform scalars -> FP SALU (s_*_f32) + s_load;
//    theta loads / output stores coalesced; outputs via nontemporal stores.
// WMMA has no profitable mapping (serial per-element complex 2-vector chain).
// ---------------------------------------------------------------------------

typedef float f2 __attribute__((ext_vector_type(2)));

__device__ __forceinline__ f2 mkc(float re, float im) { f2 r; r.x = re; r.y = im; return r; }

// complex mul: d = a*b (complex) = 2x VOP3P, no shuffles
//   t.lo = a.x*b.x            t.hi = a.x*b.y            (src0 broadcasts lo)
//   d.lo = -a.y*b.y + t.lo    d.hi =  a.y*b.x + t.hi    (src0 hi, src1 swapped)
__device__ __forceinline__ f2 cmul(f2 a, f2 b) {
  f2 t, d;
  asm("v_pk_mul_f32 %0, %1, %2 op_sel:[0,0] op_sel_hi:[0,1]"
      : "=v"(t) : "v"(a), "v"(b));
  asm("v_pk_fma_f32 %0, %1, %2, %3 op_sel:[1,1,0] op_sel_hi:[1,0,1] neg_lo:[1,0,0]"
      : "=v"(d) : "v"(a), "v"(b), "v"(t));
  return d;
}
// reciprocal of a complex number via v_rcp_f32 of |z|^2 (only used at the end)
__device__ __forceinline__ f2 crcp(f2 a) {
  float d = __builtin_amdgcn_rcpf(fmaf(a.x, a.x, a.y * a.y));
  f2 c = mkc(a.x, -a.y);
  return c * d;
}
// principal-branch complex sqrt (Re >= 0); equals cos(arcsin(z)) for 1-z^2
__device__ __forceinline__ f2 csqrtc(f2 z) {
  float m = __builtin_amdgcn_sqrtf(fmaf(z.x, z.x, z.y * z.y));
  float u = __builtin_amdgcn_sqrtf(fmaxf(0.5f * (m + z.x), 0.0f));
  float v = __builtin_amdgcn_sqrtf(fmaxf(0.5f * (m - z.x), 0.0f));
  return mkc(u, copysignf(v, z.y));
}
// p = n * cos(theta_layer) = n * sqrt(1 - (num/n)^2), num = n0*sin(theta0)
__device__ __forceinline__ f2 layer_ncos(f2 n, f2 num) {
  // 1/n is wave-uniform; keep it in plain FP ops so it can ride the FP SALU
  float dn = __builtin_amdgcn_rcpf(fmaf(n.x, n.x, n.y * n.y));
  f2 inv = mkc(n.x * dn, -(n.y * dn));
  f2 sn  = cmul(num, inv);
  f2 arg = mkc(1.0f - fmaf(sn.x, sn.x, -(sn.y * sn.y)), -2.0f * sn.x * sn.y);
  return cmul(n, csqrtc(arg));
}
// forward-angle correction: theta -> pi - theta is just p -> -p
__device__ __forceinline__ f2 fwd_correct(f2 q) {
  bool lossy = fabsf(q.y) > 1e-12f;
  bool fwd   = lossy ? (q.y > 0.0f) : (q.x > 0.0f);
  return fwd ? q : -q;
}

__global__ void __launch_bounds__(256)
tmm_kernel(const float* __restrict__ d_list,
           const float* __restrict__ n_re, const float* __restrict__ n_im,
           const float* __restrict__ n0_re, const float* __restrict__ n0_im,
           const float* __restrict__ ns_re, const float* __restrict__ ns_im,
           const float* __restrict__ theta0, const float* __restrict__ lam,
           float* __restrict__ out, int NL, int L, int Th)
{
  const int th = blockIdx.x * blockDim.x + threadIdx.x;
  const int l  = blockIdx.y;               // wave-uniform -> scalar loads below
  if (th >= Th) return;

  // hint the per-l layer table into cache (global_prefetch_b8)
  __builtin_prefetch(&n_re[l], 0, 3);
  __builtin_prefetch(&n_im[l], 0, 3);

  const float s0 = __sinf(theta0[th]);
  const float k0 = 6.28318530717958647f * __builtin_amdgcn_rcpf(lam[l]);

  const f2 n0v = mkc(n0_re[l], n0_im[l]);
  const f2 num = n0v * s0;                           // n0 * sin(theta0)

  // ---- substrate (rightmost factor's q), forward-corrected ----------------
  f2 carry = fwd_correct(layer_ncos(mkc(ns_re[l], ns_im[l]), num));
  const float Tnum = carry.x;                        // Re(n_sub cos_sub)

  // v = column 0 of the running (right-to-left) product; Q = prod(p_j)
  f2 v0 = mkc(1.0f, 0.0f), v1 = mkc(0.0f, 0.0f);
  f2 Q  = mkc(1.0f, 0.0f);

  // ---- interfaces j = NL .. 1 (right-to-left), each with propagation D_j --
  for (int j = NL; j >= 1; --j) {
    f2 nj = mkc(n_re[(j - 1) * L + l], n_im[(j - 1) * L + l]); // uniform
    f2 pc = layer_ncos(nj, num);                     // p_j = n_j cos_j

    // I~_j * v  via sum/difference identity (p=pc, q=carry)
    f2 a = v0 + v1, b = v0 - v1;                     // v_pk_add_f32
    f2 u = cmul(pc, a);
    f2 w = cmul(carry, b);
    f2 iv0 = u + w, iv1 = u - w;

    // D_j = diag(exp(-i*delta), exp(+i*delta)), delta = k0*d[j-1]*p_j
    float kd = k0 * d_list[j - 1];                   // uniform -> SALU
    float dr = kd * pc.x, di = kd * pc.y;
    float E  = __expf(di);
    float Ei = __builtin_amdgcn_rcpf(E);
    float sn, cs; __sincosf(dr, &sn, &cs);
    f2 em = mkc(cs, -sn) * E;                        // exp(-i*delta)
    f2 ep = mkc(cs,  sn) * Ei;                       // exp(+i*delta)
    v0 = cmul(em, iv0);
    v1 = cmul(ep, iv1);

    Q = cmul(Q, pc);                                 // accumulate prod(p_j)
    carry = pc;
  }

  // ---- interface 0 (incidence medium), forward-corrected ------------------
  f2 p0 = fwd_correct(layer_ncos(n0v, num));         // n0 * cos(theta0)
  const float Tden = p0.x;
  f2 a = v0 + v1, b = v0 - v1;
  f2 u = cmul(p0, a);
  f2 w = cmul(carry, b);
  f2 m00 = u + w, m10 = u - w;
  Q = cmul(Q, p0);                                   // prod over all NL+1 p_j

  // ---- outputs ------------------------------------------------------------
  f2 ia    = crcp(m00);
  f2 r_amp = cmul(m10, ia);            // r = A10/A00 (prefactors cancel)
  f2 t_amp = cmul(Q,   ia);            // t = 2^(NL+1) * prod(p_j) / A00
  t_amp *= exp2f((float)(NL + 1));
  float Rv = fmaf(r_amp.x, r_amp.x, r_amp.y * r_amp.y);
  float Tv = fmaf(t_amp.x, t_amp.x, t_amp.y * t_amp.y) * Tnum *
             __builtin_amdgcn_rcpf(Tden);
  float Av = 1.0f - Rv - Tv;

  const size_t LT  = (size_t)L * (size_t)Th;
  const size_t idx = (size_t)l * (size_t)Th + (size_t)th;
  // streaming results: nontemporal stores (TH=NT) — no reuse, keep caches
  // free for the layer tables
  __builtin_nontemporal_store(Tv, out + idx);
  __builtin_nontemporal_store(Rv, out + LT + idx);
  __builtin_nontemporal_store(Av, out + 2 * LT + idx);
  __builtin_nontemporal_store(t_amp, (f2*)(out + 3 * LT) + idx);
  __builtin_nontemporal_store(r_amp, (f2*)(out + 5 * LT) + idx);
}

extern "C" void kernel_launch(void* const* d_in, const int* in_sizes, int n_in,
                              void* d_out, int out_size, void* d_ws, size_t ws_size,
                              hipStream_t stream) {
  // setup_inputs() order:
  // 0 d_list(N) 1 n_list_re(N,L) 2 n_list_im(N,L) 3 n0_re(L) 4 n0_im(L)
  // 5 nsub_re(L) 6 nsub_im(L) 7 theta0_range(Th) 8 lambda_range(L)
  const float* d_list = (const float*)d_in[0];
  const float* n_re   = (const float*)d_in[1];
  const float* n_im   = (const float*)d_in[2];
  const float* n0_re  = (const float*)d_in[3];
  const float* n0_im  = (const float*)d_in[4];
  const float* ns_re  = (const float*)d_in[5];
  const float* ns_im  = (const float*)d_in[6];
  const float* theta0 = (const float*)d_in[7];
  const float* lam    = (const float*)d_in[8];

  const int NL = in_sizes[0];   // 20 layers
  const int Th = in_sizes[7];   // 512 angles
  const int L  = in_sizes[8];   // 2048 wavelengths

  dim3 block(256, 1, 1);                       // 8 wave32 waves / workgroup
  dim3 grid((Th + 255) / 256, L, 1);           // blockIdx.y = wavelength (uniform)
  tmm_kernel<<<grid, block, 0, stream>>>(d_list, n_re, n_im, n0_re, n0_im,
                                         ns_re, ns_im, theta0, lam,
                                         (float*)d_out, NL, L, Th);
}